// RGCN_1812476199285
// MI455X (gfx1250) — compile-verified
//
#include <hip/hip_runtime.h>
#include <hip/hip_bf16.h>

// ---------------------------------------------------------------------------
// RGCN (2-layer, single relation) for MI455X / gfx1250, wave32 + WMMA fp32.
//
//   h   = relu( seg_mean(x@W1, edges) + x@root1 + b1 )
//   out =       seg_mean(h@W2, edges) + h@root2 + b2
//
// GEMM inner loop per K-step (per wave): 1 ds_load_b64 (A tile in LDS,
// padded against bank conflicts) + 2 global_load_b128 (weights packed as
// {W[k],W[k+1],R[k],R[k+1]} per n so fragments land shuffle-free on even
// VGPR pairs) + 4 v_wmma_f32_16x16x4_f32 (W-path and root-path x 2 n-tiles
// share the A fragment).
// Aggregation: f32 global atomics into an L2-resident 25.6MB buffer.
// ---------------------------------------------------------------------------

typedef __attribute__((ext_vector_type(2))) float v2f;
typedef __attribute__((ext_vector_type(8))) float v8f;

#define NNODES 50000
#define NEDGES 800000
#define INC    256
#define HIDC   128
#define FEAT   128          // N for both layers
#define FEAT_SHIFT 7        // log2(128)
#define MAXK   256
#define KPAD   4            // LDS row pad (floats) -> stride 1040B, banks spread

// ---------------------------------------------------------------------------
// Pack weights per output column n, per k-pair: {W[k],W[k+1],R[k],R[k+1]}
// -> one b128 load yields both B fragments on even-aligned VGPR pairs.
// ---------------------------------------------------------------------------
__global__ void rgcn_pack_wr(const float* __restrict__ W,
                             const float* __restrict__ R,
                             float* __restrict__ WR, int K, int N)
{
    int i = blockIdx.x * blockDim.x + threadIdx.x;
    if (i >= K * N) return;
    const int k = i / N;
    const int n = i - k * N;
    const size_t base = ((size_t)n * (K >> 1) + (k >> 1)) * 4;
    WR[base + (k & 1)]     = W[i];   // W pair in .x/.y
    WR[base + 2 + (k & 1)] = R[i];   // R pair in .z/.w
}

// ---------------------------------------------------------------------------
// Dual GEMM: H = X@W, R = X@Rw.  N fixed = 128.
// Block: 128 thr (4 waves) = one 16-row m-tile; each wave two 16-col n-tiles.
// ---------------------------------------------------------------------------
__global__ __launch_bounds__(128) void rgcn_gemm_dual_wmma(
    const float* __restrict__ X,     // [M, K] row-major
    const float* __restrict__ WR,    // packed [N][K/2][4]
    float* __restrict__ H,           // [M, 128]
    float* __restrict__ R,           // [M, 128]
    int K)
{
    __shared__ float As[16 * (MAXK + KPAD)];

    const int tid  = threadIdx.x;
    const int lane = tid & 31;
    const int wave = tid >> 5;           // 0..3 -> n-tiles {2w, 2w+1}
    const int m0   = blockIdx.x << 4;
    const int KP   = K + KPAD;

    // ---- stage A tile (16 x K) into LDS, coalesced float4 copy ----------
    {
        const float4* __restrict__ src4 = (const float4*)(X + (size_t)m0 * K);
        const int k4  = K >> 2;          // float4s per row
        const int tot = 16 * k4;         // 1024 (K=256) or 512 (K=128)
        for (int i = tid; i < tot; i += 128) {
            const int row  = i / k4;
            const int col4 = i - row * k4;
            *(float4*)(As + row * KP + (col4 << 2)) = src4[i];
        }
    }
    __syncthreads();

    const int half = lane >> 4;          // 0: lanes 0-15, 1: lanes 16-31
    const int l15  = lane & 15;
    const int n0   = wave << 5;          // first of two adjacent 16-col tiles

    const float*  arow = As + l15 * KP;  // A row (LDS)
    const int     kh   = K >> 1;         // float4s per packed column
    const float4* bpa  = (const float4*)WR + (size_t)(n0 + l15)      * kh;
    const float4* bpb  = (const float4*)WR + (size_t)(n0 + 16 + l15) * kh;

    v8f accW0 = {0.f,0.f,0.f,0.f,0.f,0.f,0.f,0.f};
    v8f accR0 = {0.f,0.f,0.f,0.f,0.f,0.f,0.f,0.f};
    v8f accW1 = {0.f,0.f,0.f,0.f,0.f,0.f,0.f,0.f};
    v8f accR1 = {0.f,0.f,0.f,0.f,0.f,0.f,0.f,0.f};

    for (int k = 0; k < K; k += 4) {
        const int ka = k + (half << 1);          // even
        // A fragment (16x4 f32): lanes 0-15 hold {K0,K1}, 16-31 hold {K2,K3}
        v2f a;
        a.x = arow[ka];
        a.y = arow[ka + 1];                      // ds_load_b64
        // B fragments: b128 = {W[ka],W[ka+1],R[ka],R[ka+1]} -> no shuffles
        const float4 ba = bpa[ka >> 1];
        const float4 bb = bpb[ka >> 1];
        v2f bw0; bw0.x = ba.x; bw0.y = ba.y;
        v2f br0; br0.x = ba.z; br0.y = ba.w;
        v2f bw1; bw1.x = bb.x; bw1.y = bb.y;
        v2f br1; br1.x = bb.z; br1.y = bb.w;

        accW0 = __builtin_amdgcn_wmma_f32_16x16x4_f32(
                    false, a, false, bw0, (short)0, accW0, false, false);
        accR0 = __builtin_amdgcn_wmma_f32_16x16x4_f32(
                    false, a, false, br0, (short)0, accR0, false, false);
        accW1 = __builtin_amdgcn_wmma_f32_16x16x4_f32(
                    false, a, false, bw1, (short)0, accW1, false, false);
        accR1 = __builtin_amdgcn_wmma_f32_16x16x4_f32(
                    false, a, false, br1, (short)0, accR1, false, false);
    }

    // C/D layout: VGPR v holds M = m0 + v + 8*half, N = ntile_base + (lane&15)
    const int c0 = n0 + l15;
#pragma unroll
    for (int v = 0; v < 8; ++v) {
        const size_t row = (size_t)(m0 + v + (half << 3)) << FEAT_SHIFT;
        H[row + c0]      = accW0[v];
        H[row + c0 + 16] = accW1[v];
        R[row + c0]      = accR0[v];
        R[row + c0 + 16] = accR1[v];
    }
}

// ---------------------------------------------------------------------------
__global__ void rgcn_zero(float* __restrict__ p, size_t n)
{
    size_t i = (size_t)blockIdx.x * blockDim.x + threadIdx.x;
    if (i < n) p[i] = 0.0f;
}

__global__ void rgcn_degree(const int* __restrict__ dst,
                            float* __restrict__ deg, int E)
{
    int e = blockIdx.x * blockDim.x + threadIdx.x;
    if (e < E) atomicAdd(&deg[dst[e]], 1.0f);
}

__global__ void rgcn_scatter(const float* __restrict__ H,
                             const int* __restrict__ src,
                             const int* __restrict__ dst,
                             float* __restrict__ AGG, int E)
{
    size_t i = (size_t)blockIdx.x * blockDim.x + threadIdx.x;
    if (i >= (size_t)E << FEAT_SHIFT) return;
    const int e = (int)(i >> FEAT_SHIFT);
    const int c = (int)(i & (FEAT - 1));
    const int s = src[e];
    const int d = dst[e];
    atomicAdd(&AGG[((size_t)d << FEAT_SHIFT) + c],
              H[((size_t)s << FEAT_SHIFT) + c]);
}

__global__ void rgcn_finalize(const float* __restrict__ AGG,
                              const float* __restrict__ R,
                              const float* __restrict__ bias,
                              const float* __restrict__ deg,
                              float* __restrict__ out,
                              size_t total, int do_relu)
{
    size_t i = (size_t)blockIdx.x * blockDim.x + threadIdx.x;
    if (i >= total) return;
    const size_t row = i >> FEAT_SHIFT;
    const int    col = (int)(i & (FEAT - 1));
    const float  d   = fmaxf(deg[row], 1.0f);
    float v = AGG[i] / d + R[i] + bias[col];
    if (do_relu) v = fmaxf(v, 0.0f);
    out[i] = v;
}

// ---------------------------------------------------------------------------
extern "C" void kernel_launch(void* const* d_in, const int* in_sizes, int n_in,
                              void* d_out, int out_size, void* d_ws, size_t ws_size,
                              hipStream_t stream)
{
    (void)in_sizes; (void)n_in; (void)out_size; (void)ws_size;

    const float* x     = (const float*)d_in[0];
    const int*   edge  = (const int*)  d_in[1];   // [2, E]
    const float* W1    = (const float*)d_in[2];
    const float* root1 = (const float*)d_in[3];
    const float* b1    = (const float*)d_in[4];
    const float* W2    = (const float*)d_in[5];
    const float* root2 = (const float*)d_in[6];
    const float* b2    = (const float*)d_in[7];
    float*       out   = (float*)d_out;

    const int* src = edge;            // edge_index[0]
    const int* dst = edge + NEDGES;   // edge_index[1]

    // workspace partition (floats)
    const size_t NM  = (size_t)NNODES * FEAT;     // 6.4M floats per buffer
    float* bufH  = (float*)d_ws;                  // x@W   (messages)
    float* bufR  = bufH  + NM;                    // x@root
    float* bufA  = bufR  + NM;                    // aggregation accumulator
    float* bufX2 = bufA  + NM;                    // layer-1 activations h
    float* deg   = bufX2 + NM;                    // [NNODES]
    float* WR1   = deg   + NNODES;                // packed [128][128][4]
    float* WR2   = WR1   + (size_t)2 * INC  * FEAT; // packed [128][64][4]

    const int T = 256;
    const int mblocks   = NNODES / 16;                                   // 3125
    const int zblocksNM = (int)((NM + T - 1) / T);
    const int zblocksN  = (NNODES + T - 1) / T;
    const int eblocks   = (NEDGES + T - 1) / T;
    const int seblocks  = (int)((((size_t)NEDGES << FEAT_SHIFT) + T - 1) / T);
    const int pblocks1  = (INC  * FEAT + T - 1) / T;
    const int pblocks2  = (HIDC * FEAT + T - 1) / T;

    // weight packing + degrees (shared by both layers)
    rgcn_pack_wr<<<pblocks1, T, 0, stream>>>(W1, root1, WR1, INC,  FEAT);
    rgcn_pack_wr<<<pblocks2, T, 0, stream>>>(W2, root2, WR2, HIDC, FEAT);
    rgcn_zero   <<<zblocksN, T, 0, stream>>>(deg, (size_t)NNODES);
    rgcn_degree <<<eblocks,  T, 0, stream>>>(dst, deg, NEDGES);

    // ---- layer 1: h = relu( mean_agg(x@W1) + x@root1 + b1 ) --------------
    rgcn_gemm_dual_wmma<<<mblocks, 128, 0, stream>>>(x, WR1, bufH, bufR, INC);
    rgcn_zero    <<<zblocksNM, T, 0, stream>>>(bufA, NM);
    rgcn_scatter <<<seblocks,  T, 0, stream>>>(bufH, src, dst, bufA, NEDGES);
    rgcn_finalize<<<zblocksNM, T, 0, stream>>>(bufA, bufR, b1, deg, bufX2,
                                               NM, /*relu=*/1);

    // ---- layer 2: out = mean_agg(h@W2) + h@root2 + b2 --------------------
    rgcn_gemm_dual_wmma<<<mblocks, 128, 0, stream>>>(bufX2, WR2, bufH, bufR, HIDC);
    rgcn_zero    <<<zblocksNM, T, 0, stream>>>(bufA, NM);
    rgcn_scatter <<<seblocks,  T, 0, stream>>>(bufH, src, dst, bufA, NEDGES);
    rgcn_finalize<<<zblocksNM, T, 0, stream>>>(bufA, bufR, b2, deg, out,
                                               NM, /*relu=*/0);
}